// BlurDownsample_47983374631244
// MI455X (gfx1250) — compile-verified
//
#include <hip/hip_runtime.h>

// Fused depthwise 3x3 binomial blur (/16) + exact 2x2 mean pool ==
// separable 4-tap [1,3,3,1]/8 stride-2 conv per axis (1/64 total), zero padded.
// B=8, C=128, H=256, W=256 fp32  ->  out 8x128x128x128 fp32.
//
// Memory-bound: 320 MB total traffic -> ~14 us at 23.3 TB/s. Strategy:
//  - async global->LDS staging of a 34x256 input strip per workgroup
//  - horizontal [1,3,3,1] pass into an LDS h-buffer
//  - vertical pass as 9x V_WMMA_F32_16X16X4_F32 against a banded stride-2
//    coefficient matrix generated per-lane in VALU.

typedef __attribute__((ext_vector_type(2))) float v2f;
typedef __attribute__((ext_vector_type(8))) float v8f;

#define XPITCH 264   // dwords; data cols at [4..259], halo zeros at 3 and 260
#define HPITCH 136   // dwords; 2*HPITCH % 64 == 16 -> conflict-free B gather

__device__ __forceinline__ float vcoef(int d) {
  // vertical taps [1,3,3,1]/64 on d in [0,3], else 0
  if ((unsigned)d >= 4u) return 0.0f;
  return (d == 1 || d == 2) ? 0.046875f : 0.015625f;
}

__global__ __launch_bounds__(256)
void blurdown_kernel(const float* __restrict__ x, float* __restrict__ out) {
  __shared__ float xt[34 * XPITCH];   // staged input strip (35.9 KB)
  __shared__ float hs[36 * HPITCH];   // horizontally filtered rows (19.6 KB)

  const int tid   = threadIdx.x;
  const int strip = blockIdx.x;       // 0..7  : 16 output rows each
  const int bc    = blockIdx.y;       // 0..1023 : (batch, channel) image
  const int gr0   = strip * 32 - 1;   // first input row of the strip

  const float* xp = x + (size_t)bc * 65536;   // 256*256 image

  // ---- zero halo columns (global cols -1 and 256 -> LDS cols 3 and 260)
  if (tid < 34) {
    xt[tid * XPITCH + 3]   = 0.0f;
    xt[tid * XPITCH + 260] = 0.0f;
  }
  // ---- zero the one out-of-image row (conv zero padding)
  {
    int rbad = (strip == 0) ? 0 : ((strip == 7) ? 33 : -1);
    if (rbad >= 0) xt[rbad * XPITCH + 4 + tid] = 0.0f;
  }
  // ---- zero h pad rows 34,35 (K-tail of the last WMMA chunk)
  hs[(34 + (tid >> 7)) * HPITCH + (tid & 127)] = 0.0f;

  // ---- async stage: 34 rows x 1024 B, 16 B per lane, 4 rows per issue round
  const int part = tid & 63;          // 64 lanes cover one 1024 B row
  const int rofs = tid >> 6;
  #pragma unroll
  for (int k = 0; k < 9; ++k) {
    int r = k * 4 + rofs;
    if (r < 34) {
      int gr = gr0 + r;
      if (gr >= 0 && gr < 256) {
        unsigned lds = (unsigned)(size_t)&xt[r * XPITCH + 4 + part * 4]; // 16B aligned
        unsigned long long ga =
            (unsigned long long)(size_t)(xp + ((size_t)gr << 8) + (part << 2));
        asm volatile("global_load_async_to_lds_b128 %0, %1, off"
                     :: "v"(lds), "v"(ga) : "memory");
      }
    }
  }
  asm volatile("s_wait_asynccnt 0" ::: "memory");
  __syncthreads();

  // ---- horizontal pass: h(r,j) = x[2j-1..2j+2] . [1,3,3,1]   (34x128 values)
  for (int q = 0; q < 17; ++q) {
    int idx = tid + q * 256;          // 0..4351
    int r = idx >> 7;
    int j = idx & 127;
    const float* b = &xt[r * XPITCH + 2 * j + 3];
    hs[r * HPITCH + j] = b[0] + 3.0f * b[1] + 3.0f * b[2] + b[3];
  }
  __syncthreads();

  // ---- vertical pass as WMMA: out(16x16) = V(16x36) x h(36x16)
  // V[i][r] = [1,3,3,1]/64 at r-2i in [0,3]; K split into 9 chunks of 4.
  const int lane = tid & 31;
  const int wid  = tid >> 5;          // 8 waves -> 8 col tiles of 16
  const int j0   = wid << 4;
  const int m    = lane & 15;         // A: row M / B: col N
  const int kb   = (lane >> 4) << 1;  // K sub-offset per half-wave

  v8f c = {};
  #pragma unroll
  for (int kc = 0; kc < 9; ++kc) {
    int k0 = kc * 4 + kb;
    v2f a = { vcoef(k0 - 2 * m), vcoef(k0 + 1 - 2 * m) };       // banded V chunk
    v2f b = { hs[k0 * HPITCH + j0 + m], hs[(k0 + 1) * HPITCH + j0 + m] };
    c = __builtin_amdgcn_wmma_f32_16x16x4_f32(
            false, a, false, b, (short)0, c, false, false);
  }

  // ---- store C (VGPR p: M=p for lanes 0-15, M=p+8 for lanes 16-31)
  float* op = out + (size_t)bc * 16384
                  + (size_t)(strip * 16 + (lane >> 4) * 8) * 128
                  + (j0 + m);
  #pragma unroll
  for (int p = 0; p < 8; ++p) op[(size_t)p * 128] = c[p];
}

extern "C" void kernel_launch(void* const* d_in, const int* in_sizes, int n_in,
                              void* d_out, int out_size, void* d_ws, size_t ws_size,
                              hipStream_t stream) {
  const float* x = (const float*)d_in[0];
  // d_in[1] is the fixed binomial kernel; taps are hardcoded ([1,3,3,1]/8 fused).
  float* out = (float*)d_out;
  dim3 grid(8, 1024, 1);   // 8 row-strips x (B*C = 1024) images
  blurdown_kernel<<<grid, dim3(256, 1, 1), 0, stream>>>(x, out);
}